// Conv3d_1125281432057
// MI455X (gfx1250) — compile-verified
//
#include <hip/hip_runtime.h>

typedef __attribute__((ext_vector_type(2))) float v2f;
typedef __attribute__((ext_vector_type(8))) float v8f;

#define K3   27
#define PPK  131072
#define NVOX 262144
#define CIN  32
#define COUT 32
#define TILES_PER_K     (PPK / 16)          // 8192 tiles of 16 pairs
#define WAVES_PER_BLOCK 8                   // 256 threads, wave32
#define BLOCKS_PER_K    32
#define WAVES_PER_K     (WAVES_PER_BLOCK * BLOCKS_PER_K)  // 256

// out[v][c] = bias[c]; one float4 per thread (2,097,152 threads)
__global__ __launch_bounds__(256) void spconv_bias_init(float* __restrict__ out,
                                                        const float* __restrict__ bias) {
  int i = blockIdx.x * blockDim.x + threadIdx.x;
  float4 b4 = ((const float4*)bias)[i & 7];   // 8 float4 chunks per 32-ch row
  ((float4*)out)[i] = b4;
}

__global__ __launch_bounds__(256) void spconv_wmma(const float* __restrict__ feat,
                                                   const float* __restrict__ kern,
                                                   const int2* __restrict__ nb,
                                                   float* __restrict__ out) {
  const int k       = blockIdx.x / BLOCKS_PER_K;   // kernel offset 0..26
  const int blkInK  = blockIdx.x % BLOCKS_PER_K;
  const int wave    = threadIdx.x >> 5;
  const int lane    = threadIdx.x & 31;
  const int hi      = lane >> 4;                   // wave half (0/1)
  const int nlo     = lane & 15;
  const int waveInK = blkInK * WAVES_PER_BLOCK + wave;

  // ---- B fragments: kernel[k] as two 32x16 column halves, resident in VGPRs.
  // V_WMMA_F32_16X16X4_F32 B layout: VGPR0 = B[k0+2*hi][n], VGPR1 = B[k0+2*hi+1][n]
  const float* kb = kern + k * (CIN * COUT);
  v2f bfrag[2][8];
#pragma unroll
  for (int half = 0; half < 2; ++half) {
#pragma unroll
    for (int kk = 0; kk < 8; ++kk) {
      const int krow = 4 * kk + 2 * hi;
      const int col  = half * 16 + nlo;
      v2f b;
      b.x = kb[(krow + 0) * COUT + col];
      b.y = kb[(krow + 1) * COUT + col];
      bfrag[half][kk] = b;
    }
  }

  // ---- Tile loop: 16 gathered rows -> 16x32 output -> atomic scatter
  for (int t = waveInK; t < TILES_PER_K; t += WAVES_PER_K) {
    const int base = k * PPK + t * 16;
    const int2 pr  = nb[base + nlo];           // lane L owns pair (L&15)
    const float* arow = feat + (size_t)pr.x * CIN + 2 * hi;

    // A layout: lane holds A[m][4kk+2*hi .. +1] -> contiguous b64 gather
    v2f a[8];
#pragma unroll
    for (int kk = 0; kk < 8; ++kk)
      a[kk] = *(const v2f*)(arow + 4 * kk);

    v8f c0 = {};
    v8f c1 = {};
#pragma unroll
    for (int kk = 0; kk < 8; ++kk) {
      c0 = __builtin_amdgcn_wmma_f32_16x16x4_f32(false, a[kk], false, bfrag[0][kk],
                                                 (short)0, c0, false, false);
      c1 = __builtin_amdgcn_wmma_f32_16x16x4_f32(false, a[kk], false, bfrag[1][kk],
                                                 (short)0, c1, false, false);
    }

    // ---- Scatter-add: C/D VGPR r holds row (r + 8*hi), col nlo (+half*16)
#pragma unroll
    for (int r = 0; r < 8; ++r) {
      const int d = __builtin_amdgcn_ds_bpermute((r + 8 * hi) << 2, pr.y);
      float* orow = out + (size_t)d * COUT + nlo;
      unsafeAtomicAdd(orow,      c0[r]);   // no-return global_atomic_add_f32
      unsafeAtomicAdd(orow + 16, c1[r]);
    }
  }
}

extern "C" void kernel_launch(void* const* d_in, const int* in_sizes, int n_in,
                              void* d_out, int out_size, void* d_ws, size_t ws_size,
                              hipStream_t stream) {
  const float* feat = (const float*)d_in[0];   // (262144, 32) f32
  const float* kern = (const float*)d_in[1];   // (27, 32, 32) f32
  const float* bias = (const float*)d_in[2];   // (32,) f32
  const int2*  nb   = (const int2*)d_in[3];    // (M, 2) i32
  float* out = (float*)d_out;                  // (262144, 32) f32

  // 1) out = bias (also clears poison) — stream-ordered before scatter-add.
  const int n4 = (NVOX * COUT) / 4;
  spconv_bias_init<<<n4 / 256, 256, 0, stream>>>(out, bias);

  // 2) gather -> f32 WMMA -> atomic scatter-add
  spconv_wmma<<<K3 * BLOCKS_PER_K, 256, 0, stream>>>(feat, kern, nb, out);
}